// ModLayer_42795054137768
// MI455X (gfx1250) — compile-verified
//
#include <hip/hip_runtime.h>

// CDNA5 / gfx1250: wave32, WMMA 16x16x4 f32.
typedef float v2f __attribute__((ext_vector_type(2)));
typedef float v8f __attribute__((ext_vector_type(8)));

#define INV_PI_F 0.31830988618379067f
#define TWO_PI_F 6.28318530717959f

// Each wave of 32 lanes handles a tile of 16 elements.
// Lane j (0..15) and lane j+16 both own element (tileBase + j):
//   lane j     runs step-4 Chebyshev chains for n%4 == 1,2 -> B-matrix K rows 0,1
//   lane j+16  runs step-4 Chebyshev chains for n%4 == 3,0 -> B-matrix K rows 2,3
// A-matrix holds the shared 1/n weights (replicated over the 16 M-rows), so
// D = A x B + C accumulates sum_n sin(n*theta_j)/n identically into every row;
// acc[0] in every lane is the answer for its element.
__global__ __launch_bounds__(256) void modlayer_fourier_wmma(
    const float* __restrict__ x, const float* __restrict__ Kp,
    float* __restrict__ out, int B)
{
    // 1/n weight table: wtab[i] = 1/(i+1), so inv[n] = wtab[n-1]; laid out so
    // each (chunk, lane-half) weight pair is one 8-byte-aligned LDS read.
    __shared__ float wtab[1024];
    for (int i = threadIdx.x; i < 1024; i += 256)
        wtab[i] = (i < 1000) ? (1.0f / (float)(i + 1)) : 0.0f;
    __syncthreads();

    const float K   = Kp[0];
    const int lane  = threadIdx.x & 31;
    const int wave  = threadIdx.x >> 5;
    const int col   = lane & 15;   // element column within the 16-wide tile
    const int half  = lane >> 4;   // 0: K rows {0,1}; 1: K rows {2,3}

    const int e  = blockIdx.x * 128 + wave * 16 + col;
    const int el = (e < B) ? e : (B - 1);   // clamp loads; EXEC must stay all-1

    const float theta = (TWO_PI_F / K) * (x[el] + 0.5f);

    // One accurate sin/cos per element, then pure algebra.
    const float s1 = sinf(theta);
    const float c1 = cosf(theta);
    const float s2 = 2.0f * s1 * c1;
    const float c2 = 1.0f - 2.0f * s1 * s1;
    const float s3 = s1 * c2 + c1 * s2;
    const float s4 = 2.0f * s2 * c2;
    const float c4 = 1.0f - 2.0f * s2 * s2;
    const float coef = 2.0f * c4;           // 2*cos(4*theta), step-4 recurrence

    // Chain state: cur = sin(r*theta), prev = sin((r-4)*theta) at chunk 0.
    // half==0 -> r={1,2};  half==1 -> r={3,4}.
    v2f cur, prev;
    cur.x  = half ? s3   : s1;
    cur.y  = half ? s4   : s2;
    prev.x = half ? -s1  : -s3;   // sin(-1θ) / sin(-3θ)
    prev.y = half ? 0.0f : -s2;   // sin( 0θ) / sin(-2θ)

    v8f acc = {};
    const float* wbase = &wtab[2 * half];

    // Software-pipelined over chunk PAIRS (2 chunks = 8 n-terms per iteration).
    // Weights for pair p+1 are fetched while pair p's WMMAs consume the
    // registers loaded last iteration, so the wait is dscnt<=1, not 0.
    v2f a0, a1;                    // current pair's weights (chunks 2p, 2p+1)
    a0.x = wbase[0]; a0.y = wbase[1];
    a1.x = wbase[4]; a1.y = wbase[5];

#pragma unroll 2
    for (int p = 0; p < 124; ++p) {
        const int nb = 8 * p + 8;  // next pair's weight base
        v2f n0, n1;
        n0.x = wbase[nb];     n0.y = wbase[nb + 1];
        n1.x = wbase[nb + 4]; n1.y = wbase[nb + 5];

        // chunk 2p
        acc = __builtin_amdgcn_wmma_f32_16x16x4_f32(
            false, a0, false, cur, (short)0, acc, false, false);
        {
            const float t0 = __builtin_fmaf(coef, cur.x, -prev.x);
            const float t1 = __builtin_fmaf(coef, cur.y, -prev.y);
            prev = cur; cur.x = t0; cur.y = t1;
        }
        // chunk 2p+1
        acc = __builtin_amdgcn_wmma_f32_16x16x4_f32(
            false, a1, false, cur, (short)0, acc, false, false);
        {
            const float t0 = __builtin_fmaf(coef, cur.x, -prev.x);
            const float t1 = __builtin_fmaf(coef, cur.y, -prev.y);
            prev = cur; cur.x = t0; cur.y = t1;
        }
        a0 = n0; a1 = n1;
    }

    // Final pair: chunks 248, 249 (n = 993..1000)
    acc = __builtin_amdgcn_wmma_f32_16x16x4_f32(
        false, a0, false, cur, (short)0, acc, false, false);
    {
        const float t0 = __builtin_fmaf(coef, cur.x, -prev.x);
        const float t1 = __builtin_fmaf(coef, cur.y, -prev.y);
        prev = cur; cur.x = t0; cur.y = t1;
    }
    acc = __builtin_amdgcn_wmma_f32_16x16x4_f32(
        false, a1, false, cur, (short)0, acc, false, false);

    // Every row of D is identical (A rows identical) -> acc[0] is this
    // element's sum in ALL 32 lanes; lanes 16..31 are duplicates.
    const float sum = acc[0];
    const float res = __builtin_fmaf(sum, -K * INV_PI_F,
                                     __builtin_fmaf(K, 0.5f, -0.5f));
    if (half == 0 && e < B)
        out[e] = res;
}

extern "C" void kernel_launch(void* const* d_in, const int* in_sizes, int n_in,
                              void* d_out, int out_size, void* d_ws, size_t ws_size,
                              hipStream_t stream) {
    const float* x = (const float*)d_in[0];
    const float* K = (const float*)d_in[1];   // scalar parameter, 1 element
    float* out = (float*)d_out;
    const int B = in_sizes[0];

    const int elemsPerBlock = 128;            // 8 waves x 16-element tiles
    const int blocks = (B + elemsPerBlock - 1) / elemsPerBlock;
    modlayer_fourier_wmma<<<blocks, 256, 0, stream>>>(x, K, out, B);
}